// LIFLayer_37074157699151
// MI455X (gfx1250) — compile-verified
//
#include <hip/hip_runtime.h>
#include <hip/hip_bf16.h>
#include <stdint.h>

// LIF spiking recurrence, T time-major steps over a [B*N] state.
//   v = 0.5*v + x[t];  spike = v > 0.5;  out = 0.5*spike;  v = spike ? 0 : v
//
// Memory-bound streaming kernel (268 MB traffic ~= 11.5 us at 23.3 TB/s).
// vmem lives in VGPRs across all T steps; x is staged per-wave through LDS
// with the gfx1250 async copy engine (ASYNCcnt), 4-deep pipelined so each
// wave keeps 3 x 512 B in flight (~12.6 MB grid-wide, matching the HBM
// latency-bandwidth product). Outputs are non-temporal 128-bit stores.

#define LIF_BLOCK 256

typedef float v4f __attribute__((ext_vector_type(4)));

__device__ __forceinline__ unsigned lds_offset_of(const void* p) {
    // Flat shared-aperture addresses carry the LDS byte offset in bits [31:0]
    // (CDNA5 ISA 10.2).
    return (unsigned)(uintptr_t)p;
}

__device__ __forceinline__ void async_load_b128(unsigned lds_off, unsigned gbyte_off,
                                                unsigned long long sbase) {
    // global_load_async_to_lds_b128 vdst(lds addr), vaddr(byte off), saddr
    // NT hint: x is read exactly once; don't pollute the 192 MB L2 with it.
    asm volatile("global_load_async_to_lds_b128 %0, %1, %2 th:TH_LOAD_NT"
                 :
                 : "v"(lds_off), "v"(gbyte_off), "s"(sbase)
                 : "memory");
}

__device__ __forceinline__ void async_load_b128_recycle(unsigned lds_off,
                                                        unsigned gbyte_off,
                                                        unsigned long long sbase) {
    // Re-issuing into a buffer whose ds_load may still be in flight: the async
    // engine is unordered with DS ops, so fence DScnt first (the value was
    // already consumed by VALU, so this wait is ~free in steady state).
    asm volatile("s_wait_dscnt 0x0\n\t"
                 "global_load_async_to_lds_b128 %0, %1, %2 th:TH_LOAD_NT"
                 :
                 : "v"(lds_off), "v"(gbyte_off), "s"(sbase)
                 : "memory");
}

__device__ __forceinline__ void wait_async_le(int n) {
    // n is a compile-time constant after full unroll; immediates only.
    switch (n) {
        case 0:  asm volatile("s_wait_asynccnt 0x0" ::: "memory"); break;
        case 1:  asm volatile("s_wait_asynccnt 0x1" ::: "memory"); break;
        case 2:  asm volatile("s_wait_asynccnt 0x2" ::: "memory"); break;
        default: asm volatile("s_wait_asynccnt 0x3" ::: "memory"); break;
    }
}

template <int T, int BN>
__global__ __launch_bounds__(LIF_BLOCK) void lif_pipe_kernel(
    const float* __restrict__ x, float* __restrict__ out) {
    constexpr int DEPTH = 4;                      // LDS buffers; DEPTH-1 in flight
    constexpr unsigned STEP_BYTE = (unsigned)BN * 4u;

    __shared__ alignas(16) float smem[DEPTH][LIF_BLOCK * 4];

    const int tid = threadIdx.x;
    const int gid = blockIdx.x * LIF_BLOCK + tid;     // float4 index
    const unsigned elem_byte = (unsigned)gid * 16u;   // this lane's 16B chunk
    const unsigned long long xbase = (unsigned long long)(uintptr_t)x;

    unsigned lds[DEPTH];
#pragma unroll
    for (int p = 0; p < DEPTH; ++p) lds[p] = lds_offset_of(&smem[p][tid * 4]);

    // Prologue: stage t = 0..DEPTH-1.
#pragma unroll
    for (int p = 0; p < DEPTH; ++p)
        async_load_b128(lds[p], (unsigned)p * STEP_BYTE + elem_byte, xbase);

    v4f vmem = {0.0f, 0.0f, 0.0f, 0.0f};

#pragma unroll
    for (int t = 0; t < T; ++t) {
        // Oldest outstanding copy is t's: outstanding = min(DEPTH, T-t),
        // so wait until at most min(DEPTH-1, T-1-t) remain.
        wait_async_le((T - 1 - t) < (DEPTH - 1) ? (T - 1 - t) : (DEPTH - 1));

        const v4f xt = *(const v4f*)&smem[t % DEPTH][tid * 4];  // ds_load_b128

        v4f o;
#pragma unroll
        for (int c = 0; c < 4; ++c) {
            const float v = 0.5f * vmem[c] + xt[c];   // leaky integrate (fma)
            const bool  s = v > 0.5f;                 // fire (v - THRESH > 0)
            o[c]    = s ? 0.5f : 0.0f;                // spike * THRESH
            vmem[c] = s ? 0.0f : v;                   // hard reset
        }

        // Stream-once output: non-temporal 128-bit store.
        __builtin_nontemporal_store(
            o, (v4f*)(out + (size_t)t * (size_t)BN + (size_t)gid * 4));

        // Recycle the buffer just consumed for timestep t+DEPTH.
        if (t + DEPTH < T)
            async_load_b128_recycle(lds[t % DEPTH],
                                    (unsigned)(t + DEPTH) * STEP_BYTE + elem_byte,
                                    xbase);
    }
}

// Fallback for non-reference shapes: plain NT streaming, 1 float per thread.
__global__ void lif_generic_kernel(const float* __restrict__ x,
                                   float* __restrict__ out, int T, int BN) {
    const int i = blockIdx.x * blockDim.x + threadIdx.x;
    if (i >= BN) return;
    float vm = 0.0f;
    for (int t = 0; t < T; ++t) {
        const size_t idx = (size_t)t * (size_t)BN + (size_t)i;
        const float v  = 0.5f * vm + __builtin_nontemporal_load(x + idx);
        const float sp = (v > 0.5f) ? 1.0f : 0.0f;
        __builtin_nontemporal_store(0.5f * sp, out + idx);
        vm = v * (1.0f - sp);
    }
}

extern "C" void kernel_launch(void* const* d_in, const int* in_sizes, int n_in,
                              void* d_out, int out_size, void* d_ws, size_t ws_size,
                              hipStream_t stream) {
    const float* x = (const float*)d_in[0];
    float* out     = (float*)d_out;

    constexpr int T  = 32;              // fixed by the reference recurrence
    constexpr int BN = 128 * 8192;      // reference state size per timestep
    const int total  = in_sizes[0];

    if (total == T * BN) {
        constexpr int blocks = BN / (LIF_BLOCK * 4);   // 1024 blocks = 8192 waves
        lif_pipe_kernel<T, BN><<<blocks, LIF_BLOCK, 0, stream>>>(x, out);
    } else {
        const int bn = total / T;
        if (bn > 0) {
            const int blocks = (bn + LIF_BLOCK - 1) / LIF_BLOCK;
            lif_generic_kernel<<<blocks, LIF_BLOCK, 0, stream>>>(x, out, T, bn);
        }
    }
}